// AttributeDecoder_23871428231491
// MI455X (gfx1250) — compile-verified
//
#include <hip/hip_runtime.h>
#include <hip/hip_bf16.h>

typedef __attribute__((ext_vector_type(2))) float v2f;
typedef __attribute__((ext_vector_type(4))) float v4f;
typedef __attribute__((ext_vector_type(8))) float v8f;

// ---------------- degree / normalization ----------------
__global__ void fill_ones_kernel(float* __restrict__ p, int n) {
  int i = blockIdx.x * blockDim.x + threadIdx.x;
  if (i < n) p[i] = 1.0f;  // self-loop contributes 1 to every node's degree
}

__global__ void deg_accumulate_kernel(const long long* __restrict__ dst,
                                      float* __restrict__ deg, int n_edges) {
  int e = blockIdx.x * blockDim.x + threadIdx.x;
  if (e < n_edges) atomicAdd(&deg[(int)dst[e]], 1.0f);
}

__global__ void rsqrt_inplace_kernel(float* __restrict__ p, int n) {
  int i = blockIdx.x * blockDim.x + threadIdx.x;
  if (i < n) p[i] = rsqrtf(p[i]);  // deg >= 1 always (self-loops)
}

// ---------------- dense GEMM via V_WMMA_F32_16X16X4_F32 ----------------
// H[n_nodes x Ndim] = X[n_nodes x Kdim] * W[Kdim x Ndim], Kdim/Ndim compile-time.
// blockDim.x = 32 * (Ndim/16); each block computes 16 rows x Ndim cols.
// One wave32 per 16x16 output tile; K marched in steps of 4.
//
// W staged in LDS pair-interleaved: element (k, n) at sW[(k>>1)*2*Ndim + 2n + (k&1)]
// so each lane's (B[kb][n], B[kb+1][n]) pair is one contiguous 8B ds_load_b64,
// with compile-time strides (constant-folded addressing, no per-step multiplies).
template <int Kdim, int Ndim>
__global__ __launch_bounds__(256) void gemm_wmma_f32_kernel(
    const float* __restrict__ X, const float* __restrict__ W,
    float* __restrict__ H, int n_nodes) {
  __shared__ float sA[16 * Kdim];     // A tile (<= 8 KB)
  __shared__ float sW[Kdim * Ndim];   // full W, pair-interleaved (32 KB)

  constexpr int NWAVES = Ndim / 16;
  constexpr int BD     = 32 * NWAVES;
  constexpr int Ndim2  = Ndim * 2;

  const int tid  = threadIdx.x;
  const int row0 = blockIdx.x * 16;

  // Cooperative 16B global loads into LDS (Kdim % 4 == 0: float4 stays in-row).
#pragma unroll
  for (int i = tid * 4; i < 16 * Kdim; i += BD * 4) {
    int r = i / Kdim, c = i % Kdim;
    int row = row0 + r;
    v4f v = {0.0f, 0.0f, 0.0f, 0.0f};
    if (row < n_nodes) v = *(const v4f*)&X[row * Kdim + c];
    *(v4f*)&sA[i] = v;
  }
#pragma unroll
  for (int i = tid * 4; i < Kdim * Ndim; i += BD * 4) {
    v4f v = *(const v4f*)&W[i];        // row-contiguous: k fixed, n = c..c+3
    int k = i / Ndim, c = i % Ndim;
    float* base = &sW[(k >> 1) * Ndim2 + (k & 1)];
    base[(c + 0) << 1] = v.x;
    base[(c + 1) << 1] = v.y;
    base[(c + 2) << 1] = v.z;
    base[(c + 3) << 1] = v.w;
  }
  __syncthreads();

  const int wave = tid >> 5;
  const int lane = tid & 31;
  const int n0   = wave * 16;            // this wave's N-tile origin
  const int m    = lane & 15;            // A row within tile (ISA 16x4 f32 layout)
  const int nn   = lane & 15;            // B/D column within tile
  const int koff = (lane & 16) ? 2 : 0;  // lanes 16-31 carry K=2,3 of each step

  const float* aBase = &sA[m * Kdim + koff];                       // 8B aligned
  const float* bBase = &sW[((n0 + nn) << 1) + (koff >> 1) * Ndim2]; // 8B aligned

  v8f acc = {};
#pragma unroll
  for (int s = 0; s < (Kdim >> 2); ++s) {
    v2f a = *(const v2f*)&aBase[s << 2];         // A[m][kb], A[m][kb+1]
    v2f b = *(const v2f*)&bBase[s * 2 * Ndim2];  // B[kb][n], B[kb+1][n]
    acc = __builtin_amdgcn_wmma_f32_16x16x4_f32(
        /*neg_a=*/false, a, /*neg_b=*/false, b,
        /*c_mod=*/(short)0, acc, /*reuse_a=*/false, /*reuse_b=*/false);
  }

  // D layout: VGPR r -> row (r | +8 for lanes 16-31), col = lane&15
  const int mbase = row0 + ((lane & 16) ? 8 : 0);
#pragma unroll
  for (int r = 0; r < 8; ++r) {
    int row = mbase + r;
    if (row < n_nodes) H[row * Ndim + n0 + nn] = acc[r];
  }
}

// ---------------- edge scatter: out[dst] += h[src] * dinv[src]*dinv[dst] ----------------
template <int F>
__global__ __launch_bounds__(256) void scatter_edges_kernel(
    const float* __restrict__ H, const long long* __restrict__ src,
    const long long* __restrict__ dst, const float* __restrict__ dinv,
    float* __restrict__ out, int n_edges) {
  constexpr int EPB = 256 / F;           // edges per block
  const int le = threadIdx.x / F;
  const int f  = threadIdx.x & (F - 1);  // F is a power of two
  const int e  = blockIdx.x * EPB + le;
  if (e >= n_edges) return;
  const int s = (int)src[e];
  const int d = (int)dst[e];
  const float norm = dinv[s] * dinv[d];
  atomicAdd(&out[d * F + f], H[s * F + f] * norm);  // global_atomic_add_f32, L2-resident
}

// ---------------- self-loop + bias (+ReLU) finalize ----------------
template <bool RELU, int F>
__global__ void finalize_kernel(const float* __restrict__ H,
                                const float* __restrict__ dinv,
                                const float* __restrict__ bias,
                                float* __restrict__ out, int n_nodes) {
  int idx = blockIdx.x * blockDim.x + threadIdx.x;
  if (idx >= n_nodes * F) return;
  int i = idx / F, f = idx & (F - 1);
  float di = dinv[i];
  float v = out[idx] + H[idx] * di * di + bias[f];
  out[idx] = RELU ? fmaxf(v, 0.0f) : v;
}

static inline size_t align256(size_t x) { return (x + 255) & ~(size_t)255; }

extern "C" void kernel_launch(void* const* d_in, const int* in_sizes, int n_in,
                              void* d_out, int out_size, void* d_ws, size_t ws_size,
                              hipStream_t stream) {
  const float*     z  = (const float*)d_in[0];
  const long long* ei = (const long long*)d_in[1];  // int64 per reference
  const float*     W1 = (const float*)d_in[2];
  const float*     b1 = (const float*)d_in[3];
  const float*     W2 = (const float*)d_in[4];
  const float*     b2 = (const float*)d_in[5];
  float* out = (float*)d_out;

  constexpr int IN = 64, HID = 128, OUT = 64;
  const int N = in_sizes[0] / IN;   // 50000
  const int E = in_sizes[1] / 2;    // 800000
  const long long* src = ei;        // edge_index[0]
  const long long* dst = ei + E;    // edge_index[1]

  // workspace carve-out (~64.2 MB)
  char* ws = (char*)d_ws;
  float* dinv = (float*)ws;  ws += align256((size_t)N * sizeof(float));
  float* h1   = (float*)ws;  ws += align256((size_t)N * HID * sizeof(float));
  float* agg1 = (float*)ws;  ws += align256((size_t)N * HID * sizeof(float));
  float* h2   = (float*)ws;  ws += align256((size_t)N * OUT * sizeof(float));

  const int TB = 256;

  // symmetric normalization: dinv = rsqrt(1 + in-degree)
  fill_ones_kernel<<<(N + TB - 1) / TB, TB, 0, stream>>>(dinv, N);
  deg_accumulate_kernel<<<(E + TB - 1) / TB, TB, 0, stream>>>(dst, dinv, E);
  rsqrt_inplace_kernel<<<(N + TB - 1) / TB, TB, 0, stream>>>(dinv, N);

  // -------- layer 1: h1 = z @ W1 ; agg1 = A_hat @ h1 ; relu(agg1 + b1) --------
  gemm_wmma_f32_kernel<IN, HID><<<(N + 15) / 16, 32 * (HID / 16), 0, stream>>>(
      z, W1, h1, N);
  (void)hipMemsetAsync(agg1, 0, (size_t)N * HID * sizeof(float), stream);
  scatter_edges_kernel<HID><<<(E + 1) / 2, 256, 0, stream>>>(
      h1, src, dst, dinv, agg1, E);
  finalize_kernel<true, HID><<<((size_t)N * HID + TB - 1) / TB, TB, 0, stream>>>(
      h1, dinv, b1, agg1, N);

  // -------- layer 2: h2 = agg1 @ W2 ; out = A_hat @ h2 + b2 --------
  gemm_wmma_f32_kernel<HID, OUT><<<(N + 15) / 16, 32 * (OUT / 16), 0, stream>>>(
      agg1, W2, h2, N);
  (void)hipMemsetAsync(out, 0, (size_t)N * OUT * sizeof(float), stream);
  scatter_edges_kernel<OUT><<<(E + 3) / 4, 256, 0, stream>>>(
      h2, src, dst, dinv, out, E);
  finalize_kernel<false, OUT><<<((size_t)N * OUT + TB - 1) / TB, TB, 0, stream>>>(
      h2, dinv, b2, out, N);
}